// ComprehensiveGANLoss_64381559767355
// MI455X (gfx1250) — compile-verified
//
#include <hip/hip_runtime.h>
#include <hip/hip_bf16.h>
#include <math.h>

typedef __attribute__((ext_vector_type(2))) float v2f;
typedef __attribute__((ext_vector_type(8))) float v8f;

// ---------------------------------------------------------------------------
// Wave32 full reduction using V_WMMA_F32_16X16X4_F32.
// A (16x4): lane L<16 -> A[L][0]=a.x, A[L][1]=a.y ; lane L>=16 -> A[L-16][2]=a.x, A[L-16][3]=a.y
// With a = {v, 0} and B = all-ones, D[m][n] = v_m + v_{m+16} (broadcast over n).
// C/D layout: vgpr j, lane n<16 -> D[j][n]; lane n>=16 -> D[j+8][n-16].
// So   lane <16 : sum_j c[j] = sum_{m=0..7}  (v_m + v_{m+16})
//      lane>=16 : sum_j c[j] = sum_{m=8..15} (v_m + v_{m+16})
// One shfl_xor(16) add -> full 32-lane sum, broadcast to every lane.
// Requires EXEC all-ones at call site (call from fully converged code only).
// ---------------------------------------------------------------------------
__device__ __forceinline__ float wave_sum32(float v) {
    v2f a; a.x = v;    a.y = 0.0f;
    v2f b; b.x = 1.0f; b.y = 1.0f;
    v8f c = {};
    c = __builtin_amdgcn_wmma_f32_16x16x4_f32(false, a, false, b, (short)0, c, false, false);
    float s = c[0] + c[1] + c[2] + c[3] + c[4] + c[5] + c[6] + c[7];
    s += __shfl_xor(s, 16, 32);
    return s;
}

__global__ void init_kernel(float* out) { out[0] = 0.0f; }

// ---------------------------------------------------------------------------
// Feature matching: one block per row position s (128 threads, 4 floats/lane).
// For each batch b: L2-normalize real/fake rows (D=512), accumulate difference.
// Contribution = scale * sum_d ( (1/B) * sum_b (f_norm - r_norm) )^2
// scale = w / (4 * S * 512)
// ---------------------------------------------------------------------------
__global__ __launch_bounds__(128)
void fm_kernel(const float* __restrict__ real, const float* __restrict__ fake,
               float* __restrict__ out, int S, int B, float scale) {
    __shared__ float ldsR[4], ldsF[4], ldsV[4];
    const int s    = blockIdx.x;
    const int t    = threadIdx.x;        // 0..127
    const int wid  = t >> 5;
    const int lane = t & 31;

    float ax = 0.f, ay = 0.f, az = 0.f, aw = 0.f;

    for (int b = 0; b < B; ++b) {
        const size_t base = ((size_t)b * (size_t)S + (size_t)s) * 512u + (size_t)t * 4u;
        const float4 r = *(const float4*)(real + base);
        const float4 f = *(const float4*)(fake + base);

        const float pr = r.x*r.x + r.y*r.y + r.z*r.z + r.w*r.w;
        const float pf = f.x*f.x + f.y*f.y + f.z*f.z + f.w*f.w;

        const float sr = wave_sum32(pr);
        const float sf = wave_sum32(pf);
        if (lane == 0) { ldsR[wid] = sr; ldsF[wid] = sf; }
        __syncthreads();
        const float nr = ldsR[0] + ldsR[1] + ldsR[2] + ldsR[3];
        const float nf = ldsF[0] + ldsF[1] + ldsF[2] + ldsF[3];
        __syncthreads();   // protect LDS before next iteration overwrites

        const float ir = 1.0f / fmaxf(sqrtf(nr), 1e-12f);
        const float iff = 1.0f / fmaxf(sqrtf(nf), 1e-12f);

        ax += f.x * iff - r.x * ir;
        ay += f.y * iff - r.y * ir;
        az += f.z * iff - r.z * ir;
        aw += f.w * iff - r.w * ir;
    }

    const float invB = 1.0f / (float)B;
    const float dx = ax * invB, dy = ay * invB, dz = az * invB, dw = aw * invB;
    const float v  = dx*dx + dy*dy + dz*dz + dw*dw;

    const float sv = wave_sum32(v);
    if (lane == 0) ldsV[wid] = sv;
    __syncthreads();
    if (t == 0) {
        const float tot = ldsV[0] + ldsV[1] + ldsV[2] + ldsV[3];
        atomicAdd(out, tot * scale);
    }
}

// ---------------------------------------------------------------------------
// Adversarial: mean softplus(-x) * w  ==>  coef = w / n per element.
// softplus(-x) = max(-x,0) + log1p(exp(-|x|))   (numerically stable)
// ---------------------------------------------------------------------------
__global__ __launch_bounds__(256)
void adv_kernel(const float* __restrict__ x, int n, float coef, float* __restrict__ out) {
    __shared__ float lds[8];
    float local = 0.0f;
    for (int i = blockIdx.x * blockDim.x + threadIdx.x; i < n; i += gridDim.x * blockDim.x) {
        const float v = x[i];
        local += fmaxf(-v, 0.0f) + log1pf(expf(-fabsf(v)));
    }
    const float s = wave_sum32(local);
    const int wid = threadIdx.x >> 5, lane = threadIdx.x & 31;
    if (lane == 0) lds[wid] = s;
    __syncthreads();
    if (threadIdx.x == 0) {
        float tot = 0.0f;
        const int nw = blockDim.x >> 5;
        for (int i = 0; i < nw; ++i) tot += lds[i];
        atomicAdd(out, tot * coef);
    }
}

// ---------------------------------------------------------------------------
// Musical perceptual terms from tokens (B,S) int32.
//   rhythm:  mean |diff(time_shift)|          over (B, S-1)  -> /(B*(S-1))
//   harmony: sum_i mean_b(harsh) / S          -> /(B*S)
//   melody:  mean(|pitch diff| > 12)          over (B, S-1)  -> /(B*(S-1))
// gridDim.y indexes batch; grid-stride over i in [0, S-2].
// ---------------------------------------------------------------------------
__global__ __launch_bounds__(256)
void musical_kernel(const int* __restrict__ tok, int B, int S, float* __restrict__ out) {
    __shared__ float lds[8];
    const int b = blockIdx.y;
    const float c_rm = 1.0f / ((float)B * (float)(S - 1));
    const float c_h  = 1.0f / ((float)B * (float)S);

    float local = 0.0f;
    const int base = b * S;
    for (int i = blockIdx.x * blockDim.x + threadIdx.x; i < S - 1; i += gridDim.x * blockDim.x) {
        const int t0 = tok[base + i];
        const int t1 = tok[base + i + 1];

        const int ts0 = (t0 >= 256 && t0 < 768) ? 1 : 0;
        const int ts1 = (t1 >= 256 && t1 < 768) ? 1 : 0;
        const float rhythm = (ts0 != ts1) ? 1.0f : 0.0f;

        const int p0 = (t0 < 128) ? t0 : 0;
        const int p1 = (t1 < 128) ? t1 : 0;
        int iv = (p0 % 12) - (p1 % 12);
        iv = iv < 0 ? -iv : iv;
        const float harsh = (iv == 6 || iv == 11) ? 1.0f : 0.0f;

        int pd = p1 - p0;
        pd = pd < 0 ? -pd : pd;
        const float mel = (pd > 12) ? 1.0f : 0.0f;

        local += (rhythm + mel) * c_rm + harsh * c_h;
    }

    const float s = wave_sum32(local);
    const int wid = threadIdx.x >> 5, lane = threadIdx.x & 31;
    if (lane == 0) lds[wid] = s;
    __syncthreads();
    if (threadIdx.x == 0) {
        float tot = 0.0f;
        const int nw = blockDim.x >> 5;
        for (int i = 0; i < nw; ++i) tot += lds[i];
        atomicAdd(out, tot);
    }
}

extern "C" void kernel_launch(void* const* d_in, const int* in_sizes, int n_in,
                              void* d_out, int out_size, void* d_ws, size_t ws_size,
                              hipStream_t stream) {
    const float* real_local  = (const float*)d_in[0];
    const float* real_phrase = (const float*)d_in[1];
    const float* real_global = (const float*)d_in[2];
    const float* real_input  = (const float*)d_in[3];
    const float* fake_local  = (const float*)d_in[4];
    const float* fake_phrase = (const float*)d_in[5];
    const float* fake_global = (const float*)d_in[6];
    const float* fake_input  = (const float*)d_in[7];
    const float* local_log   = (const float*)d_in[8];
    const float* phrase_log  = (const float*)d_in[9];
    const float* global_log  = (const float*)d_in[10];
    const int*   tokens      = (const int*)d_in[11];
    float* out = (float*)d_out;

    const int B = 32, S = 2048, P = 128, D = 512;

    init_kernel<<<1, 1, 0, stream>>>(out);

    // feature matching: scale = w / (4 * S_feat * D)
    fm_kernel<<<S, 128, 0, stream>>>(real_local,  fake_local,  out, S, B, 0.4f / (4.0f * S * D));
    fm_kernel<<<P, 128, 0, stream>>>(real_phrase, fake_phrase, out, P, B, 0.4f / (4.0f * P * D));
    fm_kernel<<<1, 128, 0, stream>>>(real_global, fake_global, out, 1, B, 0.2f / (4.0f * 1 * D));
    fm_kernel<<<S, 128, 0, stream>>>(real_input,  fake_input,  out, S, B, 0.1f / (4.0f * S * D));

    // adversarial: coef = w / n
    adv_kernel<<<64, 256, 0, stream>>>(local_log,  B * S, 0.4f / (float)(B * S), out);
    adv_kernel<<<8,  256, 0, stream>>>(phrase_log, B * P, 0.4f / (float)(B * P), out);
    adv_kernel<<<1,  32,  0, stream>>>(global_log, B,     0.2f / (float)B,       out);

    // musical perceptual
    musical_kernel<<<dim3(4, B), 256, 0, stream>>>(tokens, B, S, out);
}